// FujiTopKRouter_71159018160283
// MI455X (gfx1250) — compile-verified
//
#include <hip/hip_runtime.h>
#include <hip/hip_bf16.h>

// ---------------------------------------------------------------------------
// FujiTopKRouter for MI455X (gfx1250, wave32)
//
// logits = x[16384,4096] @ w[64,4096]^T  via V_WMMA_F32_16X16X4_F32
// probs  = softmax(logits) ; top8 + renorm, all fused in one kernel.
//
// Roofline: 8.6 GFLOP vs 274 MB HBM traffic (AI ~ 31 F/B) -> HBM bound
// ~11.8us @ 23.3 TB/s; WMMA needed to exceed ~730 TF f32 effective.
// ---------------------------------------------------------------------------

typedef float v2f __attribute__((ext_vector_type(2)));
typedef float v4f __attribute__((ext_vector_type(4)));
typedef float v8f __attribute__((ext_vector_type(8)));

#define NUM_EXPERTS 64
#define TOP_K       8
#define HIDDEN      4096
#define TOKENS      16384

#define WAVES_PER_BLOCK 8
#define ROW_STRIDE      68   // 64 + 4 pad: bank = (4*row + col) & 63 -> conflict-free

__global__ __launch_bounds__(256) void fuji_topk_router_kernel(
    const float* __restrict__ x,      // [TOKENS, HIDDEN]
    const float* __restrict__ w,      // [NUM_EXPERTS, HIDDEN]
    float* __restrict__ probs,        // [TOKENS, NUM_EXPERTS]
    float* __restrict__ topv,         // [TOKENS, TOP_K]
    int*   __restrict__ topi)         // [TOKENS, TOP_K]
{
    __shared__ float lds[WAVES_PER_BLOCK][16 * ROW_STRIDE];
    __shared__ float rowinv[WAVES_PER_BLOCK][16];

    const int wid  = threadIdx.x >> 5;
    const int lane = threadIdx.x & 31;
    const int tile = blockIdx.x * WAVES_PER_BLOCK + wid;   // one 16-token tile / wave
    const int tok0 = tile * 16;

    const int m  = lane & 15;   // A: row M within tile ; B: expert column N within n-tile
    const int kh = lane >> 4;   // K-half select per ISA 16x4 f32 operand layout

    // ---------------- GEMM: 16 tokens x 64 experts, K = 4096 ----------------
    const float* aptr = x + (size_t)(tok0 + m) * HIDDEN + 4 * kh;
    const float* bptr[4];
#pragma unroll
    for (int nt = 0; nt < 4; ++nt)
        bptr[nt] = w + (size_t)(nt * 16 + m) * HIDDEN + 4 * kh;

    v8f acc[4];
#pragma unroll
    for (int nt = 0; nt < 4; ++nt) acc[nt] = (v8f)0.0f;

    // K-step of 8: lane loads float4 at memory K = k + 4*kh.
    // WMMA #0 consumes (.x,.y), WMMA #1 consumes (.z,.w); the induced logical-K
    // permutation is applied identically to A and B, so the dot product is exact.
    for (int k = 0; k < HIDDEN; k += 8) {
        v4f a = *(const v4f*)(aptr + k);
        v2f a0 = {a.x, a.y};
        v2f a1 = {a.z, a.w};
#pragma unroll
        for (int nt = 0; nt < 4; ++nt) {
            v4f b = *(const v4f*)(bptr[nt] + k);
            v2f b0 = {b.x, b.y};
            v2f b1 = {b.z, b.w};
            acc[nt] = __builtin_amdgcn_wmma_f32_16x16x4_f32(
                false, a0, false, b0, (short)0, acc[nt], false, false);
            acc[nt] = __builtin_amdgcn_wmma_f32_16x16x4_f32(
                false, a1, false, b1, (short)0, acc[nt], false, false);
        }
    }

    // ---------------- spill logits tile to LDS ----------------
    // C/D layout: VGPR i holds row M = i + 8*(lane>>4), col N = lane&15.
    float* tlds = lds[wid];
#pragma unroll
    for (int nt = 0; nt < 4; ++nt) {
#pragma unroll
        for (int i = 0; i < 8; ++i)
            tlds[(i + 8 * kh) * ROW_STRIDE + nt * 16 + m] = acc[nt][i];
    }

    // ---------------- softmax + top-8 (one token per lane 0..15) ------------
    if (lane < 16) {
        float* row = tlds + lane * ROW_STRIDE;

        float mx = row[0];
#pragma unroll
        for (int j = 1; j < NUM_EXPERTS; ++j) mx = fmaxf(mx, row[j]);

        float sum = 0.0f;
#pragma unroll
        for (int j = 0; j < NUM_EXPERTS; ++j) {
            float e = __expf(row[j] - mx);
            row[j] = e;              // keep unnormalized exp in LDS
            sum += e;
        }
        rowinv[wid][lane] = 1.0f / sum;

        // 8-pass masked argmax: descending values, first-index tie-break,
        // matching jax.lax.top_k.
        unsigned long long used = 0ull;
        float tv[TOP_K];
        int   ti[TOP_K];
        float tsum = 0.0f;
#pragma unroll
        for (int s = 0; s < TOP_K; ++s) {
            float best = -1.0f;      // exp() > 0 always
            int   bi   = 0;
            for (int j = 0; j < NUM_EXPERTS; ++j) {
                float e = row[j];
                if (!((used >> j) & 1ull) && e > best) { best = e; bi = j; }
            }
            used |= 1ull << bi;
            tv[s] = best;
            ti[s] = bi;
            tsum += best;
        }
        const float tinv = 1.0f / tsum;   // renorm over the top-8 only
        const int tok = tok0 + lane;
#pragma unroll
        for (int s = 0; s < TOP_K; ++s) {
            topv[(size_t)tok * TOP_K + s] = tv[s] * tinv;
            topi[(size_t)tok * TOP_K + s] = ti[s];
        }
    }
    // wave32: lanes reconverge; same-wave LDS ops are program-ordered.

    // ---------------- coalesced normalized-prob store ----------------
#pragma unroll
    for (int r = 0; r < 16; ++r) {
        const float inv  = rowinv[wid][r];
        const size_t base = (size_t)(tok0 + r) * NUM_EXPERTS;
        probs[base + lane]      = tlds[r * ROW_STRIDE + lane]      * inv;
        probs[base + lane + 32] = tlds[r * ROW_STRIDE + lane + 32] * inv;
    }
}

extern "C" void kernel_launch(void* const* d_in, const int* in_sizes, int n_in,
                              void* d_out, int out_size, void* d_ws, size_t ws_size,
                              hipStream_t stream) {
    const float* x = (const float*)d_in[0];   // hidden_states [16384,4096] f32
    const float* w = (const float*)d_in[1];   // router weight [64,4096] f32

    // d_out = probs [16384*64] f32 | top_value [16384*8] f32 | indices [16384*8] i32
    float* probs = (float*)d_out;
    float* topv  = probs + (size_t)TOKENS * NUM_EXPERTS;
    int*   topi  = (int*)(topv + (size_t)TOKENS * TOP_K);

    const int tiles  = TOKENS / 16;               // 1024
    const int blocks = tiles / WAVES_PER_BLOCK;   // 128 blocks x 256 threads
    fuji_topk_router_kernel<<<blocks, 256, 0, stream>>>(x, w, probs, topv, topi);
}